// MaxIndexLinearTraining_31576599561038
// MI455X (gfx1250) — compile-verified
//
#include <hip/hip_runtime.h>
#include <cstdint>

// out[b,g] = sum_q softmax(x[b,g,:])_q * (q+1),  q in 0..3
// HBM-bound streaming op: 128 MiB read + 32 MiB write / 23.3 TB/s ~= 7 us.
// No matmul structure -> no WMMA. CDNA5-specific path: async global->LDS
// double-buffered tiles (ASYNCcnt) + split waits + NT cache policy, since
// input/output are touched exactly once.

typedef float v4f __attribute__((ext_vector_type(4)));

#define TPB 256
#define GROUPS_PER_THREAD 4
#define TILE_GROUPS (TPB * GROUPS_PER_THREAD)   // 1024 groups per tile
#define TILE_BYTES  (TILE_GROUPS * 16)          // 16384 bytes per tile
#define THREAD_BYTES (GROUPS_PER_THREAD * 16)   // 64 bytes per thread per tile
#define NBUF 2

__device__ __forceinline__ float softargmax4(float x0, float x1, float x2, float x3) {
  float m  = fmaxf(fmaxf(x0, x1), fmaxf(x2, x3));
  float e0 = __expf(x0 - m);
  float e1 = __expf(x1 - m);
  float e2 = __expf(x2 - m);
  float e3 = __expf(x3 - m);
  float s  = (e0 + e1) + (e2 + e3);
  float ws = fmaf(3.0f, e3, fmaf(2.0f, e2, e1));   // 0*e0 + 1*e1 + 2*e2 + 3*e3
  return 1.0f + ws / s;                             // weights [1,2,3,4]
}

__device__ __forceinline__ void issue_tile_async(unsigned lds_base_byte,
                                                 unsigned long long gaddr) {
  // IOFFSET offsets both the global and the LDS address, so one LDS-addr VGPR
  // and one 64-bit global-addr pair cover all four b128 transfers (64 B/lane).
  // Input is read exactly once -> non-temporal.
  asm volatile("global_load_async_to_lds_b128 %0, %1, off th:TH_LOAD_NT\n\t"
               "global_load_async_to_lds_b128 %0, %1, off offset:16 th:TH_LOAD_NT\n\t"
               "global_load_async_to_lds_b128 %0, %1, off offset:32 th:TH_LOAD_NT\n\t"
               "global_load_async_to_lds_b128 %0, %1, off offset:48 th:TH_LOAD_NT"
               :: "v"(lds_base_byte), "v"(gaddr) : "memory");
}

__global__ __launch_bounds__(TPB)
void softargmax_stream(const float* __restrict__ in, float* __restrict__ out,
                       long long n_groups, long long n_tiles) {
  const int tid = threadIdx.x;
  const unsigned lane_lds = (unsigned)(tid * THREAD_BYTES);
  const unsigned long long in_base = (unsigned long long)(uintptr_t)in;
  const long long stride = gridDim.x;

  long long tile = blockIdx.x;

  // Prologue: kick off async copy of our first tile into LDS buffer 0.
  if (tile < n_tiles) {
    issue_tile_async(lane_lds,
                     in_base + (unsigned long long)tile * TILE_BYTES + lane_lds);
  }

  int buf = 0;
  while (tile < n_tiles) {
    long long next = tile + stride;
    if (next < n_tiles) {
      // Prefetch next tile into the other buffer, then wait until only the
      // prefetch (4 async ops) is outstanding -> current tile is resident.
      issue_tile_async((unsigned)((buf ^ 1) * TILE_BYTES) + lane_lds,
                       in_base + (unsigned long long)next * TILE_BYTES + lane_lds);
      asm volatile("s_wait_asynccnt 0x4" ::: "memory");
    } else {
      asm volatile("s_wait_asynccnt 0x0" ::: "memory");
    }

    // Read our 64 B back from LDS. Fuse the dscnt wait into the same asm so the
    // destination VGPRs cannot be consumed before the data has landed.
    v4f a, b, c, d;
    {
      unsigned l = (unsigned)(buf * TILE_BYTES) + lane_lds;
      asm volatile("ds_load_b128 %0, %4\n\t"
                   "ds_load_b128 %1, %4 offset:16\n\t"
                   "ds_load_b128 %2, %4 offset:32\n\t"
                   "ds_load_b128 %3, %4 offset:48\n\t"
                   "s_wait_dscnt 0x0"
                   : "=&v"(a), "=&v"(b), "=&v"(c), "=&v"(d)
                   : "v"(l)
                   : "memory");
    }

    v4f r;
    r.x = softargmax4(a.x, a.y, a.z, a.w);
    r.y = softargmax4(b.x, b.y, b.z, b.w);
    r.z = softargmax4(c.x, c.y, c.z, c.w);
    r.w = softargmax4(d.x, d.y, d.z, d.w);

    // Output written exactly once -> non-temporal 128-bit store.
    __builtin_nontemporal_store(
        r, reinterpret_cast<v4f*>(out + tile * TILE_GROUPS + tid * GROUPS_PER_THREAD));

    buf ^= 1;
    tile = next;
  }

  // Tail: any groups not covered by whole tiles (zero for the reference shape).
  for (long long gi = n_tiles * TILE_GROUPS + (long long)blockIdx.x * TPB + tid;
       gi < n_groups;
       gi += stride * TPB) {
    v4f x = reinterpret_cast<const v4f*>(in)[gi];
    out[gi] = softargmax4(x.x, x.y, x.z, x.w);
  }
}

extern "C" void kernel_launch(void* const* d_in, const int* in_sizes, int n_in,
                              void* d_out, int out_size, void* d_ws, size_t ws_size,
                              hipStream_t stream) {
  (void)n_in; (void)out_size; (void)d_ws; (void)ws_size;
  const float* in = (const float*)d_in[0];
  float* out = (float*)d_out;

  long long n_elems  = in_sizes[0];        // 4096*8192
  long long n_groups = n_elems / 4;        // Q = 4
  long long n_tiles  = n_groups / TILE_GROUPS;

  int blocks = 2048;
  if (n_tiles > 0 && n_tiles < (long long)blocks) blocks = (int)n_tiles;
  if (blocks < 1) blocks = 1;

  hipLaunchKernelGGL(softargmax_stream, dim3(blocks), dim3(TPB),
                     NBUF * TILE_BYTES /* dynamic LDS: 32 KiB double buffer */,
                     stream, in, out, n_groups, n_tiles);
}